// Student_LightGCN_73890617360947
// MI455X (gfx1250) — compile-verified
//
#include <hip/hip_runtime.h>
#include <hip/hip_bf16.h>

// LightGCN propagation for MI455X (gfx1250, wave32).
//
// Build CSR once per launch (histogram + scan + cursor scatter), then run
// three atomic-free SpMM layers: one wave per output row, neighbor rows
// staged through a 4-deep pipeline of CDNA5 async global->LDS copies
// (ASYNCcnt), edge indices chunk-preloaded and distributed via wave32
// shuffles, accumulation in registers, acc+= and final /4 fused into the
// epilogue. All hot data (38.4MB tables + 40MB CSR) is L2-resident.

#define N_USERS 100000
#define N_ITEMS 50000
#define N_NODES (N_USERS + N_ITEMS)
#define EMBED   64
#define TOTAL   (N_NODES * EMBED)   // 9,600,000 floats

// ---------------------------------------------------------------- elementwise

__global__ void lgcn_init_kernel(const float* __restrict__ u,
                                 const float* __restrict__ up,
                                 const float* __restrict__ it,
                                 const float* __restrict__ ip,
                                 float* __restrict__ ego,
                                 float* __restrict__ acc) {
    int idx = blockIdx.x * blockDim.x + threadIdx.x;
    if (idx >= TOTAL) return;
    const int uElems = N_USERS * EMBED;
    float v;
    if (idx < uElems) {
        v = u[idx] + up[idx];
    } else {
        int j = idx - uElems;
        v = it[j] + ip[j];
    }
    ego[idx] = v;
    acc[idx] = v;
}

// ---------------------------------------------------------------- CSR build

__global__ void lgcn_zero_cnt_kernel(int* __restrict__ cnt, int n) {
    int i = blockIdx.x * blockDim.x + threadIdx.x;
    if (i < n) cnt[i] = 0;
}

__global__ void lgcn_hist_kernel(const int* __restrict__ rows,
                                 int* __restrict__ cnt, int nnz) {
    int i = blockIdx.x * blockDim.x + threadIdx.x;
    if (i < nnz) atomicAdd(&cnt[rows[i]], 1);
}

// Single-workgroup chunked exclusive scan (1024 threads = 32 wave32 waves).
#define SCAN_BLOCK 1024
__global__ __launch_bounds__(SCAN_BLOCK)
void lgcn_scan_kernel(const int* __restrict__ cnt,
                      int* __restrict__ off,
                      int* __restrict__ cursor, int n) {
    __shared__ int warpsums[32];
    __shared__ int s_running;
    const int tid  = threadIdx.x;
    const int lane = tid & 31;
    const int wid  = tid >> 5;
    if (tid == 0) s_running = 0;
    __syncthreads();

    for (int base = 0; base < n; base += SCAN_BLOCK) {
        int i = base + tid;
        int v = (i < n) ? cnt[i] : 0;

        int x = v;                       // inclusive scan within the wave
        #pragma unroll
        for (int d = 1; d < 32; d <<= 1) {
            int t = __shfl_up(x, d, 32);
            if (lane >= d) x += t;
        }
        if (lane == 31) warpsums[wid] = x;
        __syncthreads();

        if (wid == 0) {                  // scan the 32 wave sums
            int w = warpsums[lane];
            #pragma unroll
            for (int d = 1; d < 32; d <<= 1) {
                int t = __shfl_up(w, d, 32);
                if (lane >= d) w += t;
            }
            warpsums[lane] = w;
        }
        __syncthreads();

        int waveoff = (wid > 0) ? warpsums[wid - 1] : 0;
        int incl = x + waveoff;
        int excl = incl - v;
        int run  = s_running;
        if (i < n) { off[i] = run + excl; cursor[i] = run + excl; }
        __syncthreads();
        if (tid == SCAN_BLOCK - 1) s_running = run + incl;
        __syncthreads();
    }
    if (tid == 0) off[n] = s_running;
}

__global__ void lgcn_scatter_kernel(const int*   __restrict__ rows,
                                    const int*   __restrict__ cols,
                                    const float* __restrict__ vals,
                                    int*   __restrict__ cursor,
                                    int*   __restrict__ ccol,
                                    float* __restrict__ cval, int nnz) {
    int i = blockIdx.x * blockDim.x + threadIdx.x;
    if (i < nnz) {
        int r = rows[i];
        int pos = atomicAdd(&cursor[r], 1);
        ccol[pos] = cols[i];
        cval[pos] = vals[i];
    }
}

// ---------------------------------------------------------------- CSR SpMM
//
// One wave per output row. Edge metadata is preloaded 32 edges at a time
// (one coalesced load pair per lane) and redistributed with __shfl; pad
// edges carry v=0 so the inner loop is branchless with full EXEC. Each
// iteration stages TWO neighbor rows (512B) into LDS via a 4-deep
// global_load_async_to_lds_b128 pipeline; all 32 lanes accumulate their
// private float2 slice. Epilogue fuses acc+= and the final *0.25.

#define SPMM_BLOCK 256
#define SPMM_WAVES 8
#define SPMM_GRID  4096
#define STAGES     4

__global__ __launch_bounds__(SPMM_BLOCK)
void lgcn_spmm_csr_kernel(const int*   __restrict__ off,
                          const int*   __restrict__ ccol,
                          const float* __restrict__ cval,
                          const float* __restrict__ x,
                          float*       __restrict__ y,     // new ego
                          float*       __restrict__ acc,   // fused +=, *scale
                          int nrows, int write_ego, float postscale) {
    // [wave][stage(4)][edge-slot(2)][64 floats] -> 16 KB
    __shared__ float stagebuf[SPMM_WAVES * STAGES * 2 * EMBED];

    const int lane = threadIdx.x & 31;
    const int wave = threadIdx.x >> 5;
    const int half = lane >> 4;
    const int sub  = lane & 15;

    const unsigned ldsbase = (unsigned)(size_t)(&stagebuf[0]);
    const unsigned wbase   = (unsigned)(wave * STAGES * 2 * EMBED);  // floats

    long r            = (long)blockIdx.x * SPMM_WAVES + wave;        // uniform
    const long stride = (long)gridDim.x * SPMM_WAVES;

    for (; r < (long)nrows; r += stride) {
        const int jb = off[r];
        const int je = off[r + 1];
        float2 sum = make_float2(0.0f, 0.0f);

        for (int cb = jb; cb < je; cb += 32) {       // chunk of <=32 edges
            int jc  = cb + lane;
            int jcl = (jc < je) ? jc : (je - 1);
            int   c_l = ccol[jcl];                          // coalesced
            float v_l = (jc < je) ? cval[jc] : 0.0f;        // pad -> v=0
            int nedge = je - cb; if (nedge > 32) nedge = 32;
            const int npair = (nedge + 1) >> 1;

            auto issue = [&](int k) {
                int kk = (k < npair) ? k : (npair - 1);     // clamp -> uniform cnt
                int s  = k & (STAGES - 1);
                int c  = __shfl(c_l, 2 * kk + half, 32);    // pad: valid addr, v=0
                const float* gp = x + ((long)c * EMBED + sub * 4);
                unsigned fidx  = wbase + (unsigned)(s * 2 * EMBED + half * EMBED + sub * 4);
                unsigned laddr = ldsbase + fidx * 4u;
                unsigned long long ga = (unsigned long long)(size_t)gp;
                asm volatile("global_load_async_to_lds_b128 %0, %1, off"
                             :: "v"(laddr), "v"(ga)
                             : "memory");
            };

            issue(0); issue(1); issue(2);                   // prologue (depth 4)
            for (int k = 0; k < npair; ++k) {
                issue(k + 3);
                asm volatile("s_wait_asynccnt 0x3" ::: "memory");
                const int s = k & (STAGES - 1);
                float v0 = __shfl(v_l, 2 * k,     32);
                float v1 = __shfl(v_l, 2 * k + 1, 32);
                float2 x0 = *(const float2*)(&stagebuf[wbase + s * 128 + lane * 2]);
                float2 x1 = *(const float2*)(&stagebuf[wbase + s * 128 + 64 + lane * 2]);
                sum.x += x0.x * v0;
                sum.y += x0.y * v0;
                sum.x += x1.x * v1;
                sum.y += x1.y * v1;
            }
            // drain before stage slots are reused (next chunk / next row)
            asm volatile("s_wait_asynccnt 0x0" ::: "memory");
        }

        const long o = r * EMBED + lane * 2;
        float2 a = *(const float2*)(&acc[o]);
        a.x = (a.x + sum.x) * postscale;
        a.y = (a.y + sum.y) * postscale;
        *(float2*)(&acc[o]) = a;                            // global_store_b64
        if (write_ego) {
            *(float2*)(&y[o]) = sum;
        }
    }
}

// ---------------------------------------------------------------- launcher

extern "C" void kernel_launch(void* const* d_in, const int* in_sizes, int n_in,
                              void* d_out, int out_size, void* d_ws, size_t ws_size,
                              hipStream_t stream) {
    const int*   rows = (const int*)  d_in[0];
    const int*   cols = (const int*)  d_in[1];
    const float* vals = (const float*)d_in[2];
    const float* u    = (const float*)d_in[3];
    const float* it   = (const float*)d_in[4];
    const float* up   = (const float*)d_in[5];
    const float* ip   = (const float*)d_in[6];
    float* acc = (float*)d_out;
    const int nnz = in_sizes[0];

    // workspace carve-up (all 16B-aligned)
    char* ws = (char*)d_ws;
    float* A      = (float*)ws;                 ws += (size_t)TOTAL * 4;      // 38.4MB
    float* B      = (float*)ws;                 ws += (size_t)TOTAL * 4;      // 38.4MB
    int*   cnt    = (int*)ws;                   ws += (size_t)150016 * 4;
    int*   off    = (int*)ws;                   ws += (size_t)150016 * 4;     // N_NODES+1
    int*   cursor = (int*)ws;                   ws += (size_t)150016 * 4;
    int*   ccol   = (int*)ws;                   ws += (size_t)nnz * 4;        // 19.2MB
    float* cval   = (float*)ws;                                               // 19.2MB

    const int blk   = 256;
    const int gridE = (TOTAL + blk - 1) / blk;
    const int gridN = (N_NODES + blk - 1) / blk;
    const int gridZ = (nnz + blk - 1) / blk;

    // ego/acc init + CSR build (once; adjacency reused for all 3 layers)
    lgcn_init_kernel<<<gridE, blk, 0, stream>>>(u, up, it, ip, A, acc);
    lgcn_zero_cnt_kernel<<<gridN, blk, 0, stream>>>(cnt, N_NODES);
    lgcn_hist_kernel<<<gridZ, blk, 0, stream>>>(rows, cnt, nnz);
    lgcn_scan_kernel<<<1, SCAN_BLOCK, 0, stream>>>(cnt, off, cursor, N_NODES);
    lgcn_scatter_kernel<<<gridZ, blk, 0, stream>>>(rows, cols, vals, cursor, ccol, cval, nnz);

    // 3 atomic-free SpMM layers; last one fuses the /4 and skips ego store
    for (int layer = 0; layer < 3; ++layer) {
        const int   write_ego = (layer < 2) ? 1 : 0;
        const float ps        = (layer == 2) ? 0.25f : 1.0f;
        lgcn_spmm_csr_kernel<<<SPMM_GRID, SPMM_BLOCK, 0, stream>>>(
            off, ccol, cval, A, B, acc, N_NODES, write_ego, ps);
        float* t = A; A = B; B = t;             // ping-pong ego buffers
    }
}